// SolarSpringAttention_54838142435550
// MI455X (gfx1250) — compile-verified
//
#include <hip/hip_runtime.h>
#include <math.h>

typedef __attribute__((ext_vector_type(2))) float v2f;
typedef __attribute__((ext_vector_type(8))) float v8f;
typedef __attribute__((ext_vector_type(4))) unsigned int u32x4;
typedef __attribute__((ext_vector_type(8))) unsigned int u32x8;

#define N_POS 8
#define EVENT_HORIZON 0.1f

__constant__ float POS_W_c[N_POS] = {0.95f, 0.9f, 0.85f, 0.5f, 0.4f, 0.2f, 0.15f, 0.05f};

__device__ __forceinline__ float sigmoidf(float x) { return 1.0f / (1.0f + __expf(-x)); }

// ---------------------------------------------------------------------------
// TDM helpers: 2D tile DMA global->LDS (CDNA5 Tensor Data Mover).
// D# group0 (ISA 8.3): [1:0]=count=1, [63:32]=lds_addr, [120:64]=global_addr,
//                      [127:126]=type=2.
// D# group1 (ISA 8.4): [17:16]=data_size(2=4B), [20]=pad_en, [24:22]=pad_intvl,
//                      [31:25]=pad_amt, [79:48]=tensor_dim0, [111:80]=tensor_dim1,
//                      [127:112]=tile_dim0, [143:128]=tile_dim1,
//                      [207:160]=tensor_dim0_stride.
// ---------------------------------------------------------------------------
__device__ __forceinline__ u32x4 tdm_g0(unsigned lds_byte, unsigned long long gaddr) {
    u32x4 g;
    g[0] = 1u;  // count=1, user mode, no gather
    g[1] = lds_byte;
    g[2] = (unsigned)(gaddr & 0xffffffffull);
    g[3] = (unsigned)((gaddr >> 32) & 0x01ffffffull) | 0x80000000u;  // type=2
    return g;
}

__device__ __forceinline__ u32x8 tdm_g1(unsigned tensor_d0, unsigned tensor_d1,
                                        unsigned tile_d0, unsigned tile_d1,
                                        unsigned long long stride0,
                                        unsigned pad_en, unsigned pad_intvl,
                                        unsigned pad_amt) {
    u32x8 g;
    unsigned d0 = (2u << 16);  // data_size = 4 bytes
    if (pad_en) d0 |= (1u << 20) | (pad_intvl << 22) | (pad_amt << 25);
    g[0] = d0;
    g[1] = (tensor_d0 & 0xffffu) << 16;
    g[2] = ((tensor_d0 >> 16) & 0xffffu) | ((tensor_d1 & 0xffffu) << 16);
    g[3] = ((tensor_d1 >> 16) & 0xffffu) | (tile_d0 << 16);
    g[4] = tile_d1;  // tile_dim2 = 0 (2D)
    g[5] = (unsigned)(stride0 & 0xffffffffull);
    g[6] = (unsigned)((stride0 >> 32) & 0xffffull);  // tensor_dim1_stride = 0
    g[7] = 0u;
    return g;
}

__device__ __forceinline__ void tdm_load_2d(u32x4 g0, u32x8 g1) {
    asm volatile("tensor_load_to_lds %0, %1" ::"s"(g0), "s"(g1) : "memory");
}

__device__ __forceinline__ void tdm_wait() {
    __builtin_amdgcn_s_wait_tensorcnt(0);
}

// ---------------------------------------------------------------------------
// Kernel 1: per-token stats. One block (256 thr) per token row.
// ---------------------------------------------------------------------------
__global__ void token_stats_kernel(const float* __restrict__ X,
                                   const int* __restrict__ pos_idx,
                                   const float* __restrict__ conf,
                                   const float* __restrict__ res,
                                   const float* __restrict__ kspr,
                                   float* __restrict__ masses,
                                   float* __restrict__ bh,
                                   float* __restrict__ kpi,
                                   float* __restrict__ coll,
                                   int d) {
    int i = blockIdx.x;
    const float* row = X + (size_t)i * d;
    float s = 0.f;
    for (int k = threadIdx.x; k < d; k += blockDim.x) {
        float v = row[k];
        s += v * v;
    }
    __shared__ float red[256];
    red[threadIdx.x] = s;
    __syncthreads();
    for (int off = 128; off > 0; off >>= 1) {
        if ((int)threadIdx.x < off) red[threadIdx.x] += red[threadIdx.x + off];
        __syncthreads();
    }
    if (threadIdx.x == 0) {
        int p = ((pos_idx[i] % N_POS) + N_POS) % N_POS;
        float m = sqrtf(red[0]) * POS_W_c[p] * (1.f + res[i]);
        float c = conf[i];
        float gap = fmaxf(c - EVENT_HORIZON, 1e-6f);
        float b = -0.01f / (gap * gap);
        float cl = (c <= EVENT_HORIZON) ? 1.f : 0.f;
        if (cl > 0.f) b = -1e6f;
        masses[i] = m;
        bh[i] = b;
        coll[i] = cl;
        kpi[i] = sigmoidf(kspr[p]);
    }
}

// ---------------------------------------------------------------------------
// Kernel 2: scores row + softmax row. One block (256 thr) per row i.
// Score row lives in LDS (L*4 = 16KB) so softmax never re-reads HBM.
// ---------------------------------------------------------------------------
__global__ void scores_softmax_kernel(const float* __restrict__ masses,
                                      const float* __restrict__ bh,
                                      const float* __restrict__ kpi,
                                      const float* __restrict__ coll,
                                      const int* __restrict__ pos_idx,
                                      const int* __restrict__ depth,
                                      const int* __restrict__ slot,
                                      const float* __restrict__ tpos,
                                      const float* __restrict__ G_micro,
                                      const float* __restrict__ G_macro,
                                      const float* __restrict__ temperature,
                                      float* __restrict__ scores_out,
                                      float* __restrict__ A_out,
                                      int L) {
    extern __shared__ float srow[];  // L floats
    __shared__ float sigG[64];
    __shared__ float red[256];
    int t = threadIdx.x;
    if (t < 64) sigG[t] = sigmoidf(G_micro[t]);
    int i = blockIdx.x;
    float mi = masses[i];
    float bhi = bh[i];
    float ki = kpi[i];
    int pi = ((pos_idx[i] % N_POS) + N_POS) % N_POS;
    float di = (float)depth[i];
    float si = (float)slot[i];
    float ti = tpos[i];
    float colli = coll[i];
    float gmac = sigmoidf(G_macro[0]);
    float T = fmaxf(fabsf(temperature[0]), 0.1f);
    __syncthreads();

    float lmax = -3.4e38f;
    for (int j = t; j < L; j += blockDim.x) {
        int pj = ((pos_idx[j] % N_POS) + N_POS) % N_POS;
        float mm = mi * masses[j];
        float sd = fmaxf(fabsf(si - (float)slot[j]), 1.f);
        float fmic = sigG[pi * 8 + pj] * mm / (sd * sd);
        float dd = fabsf(di - (float)depth[j]) + 1.f;
        float fmac = gmac * mm / (dd * dd);
        float fspr = ki * fabsf(ti - tpos[j]);
        float s = fmic + fmac + fspr + bhi + bh[j];
        if (colli > 0.f) s = -1e6f;
        if (j == i) s = 0.f;
        srow[j] = s;
        scores_out[(size_t)i * L + j] = s;
        lmax = fmaxf(lmax, s);
    }
    red[t] = lmax;
    __syncthreads();
    for (int off = 128; off > 0; off >>= 1) {
        if (t < off) red[t] = fmaxf(red[t], red[t + off]);
        __syncthreads();
    }
    float smax = red[0];
    __syncthreads();
    float lsum = 0.f;
    for (int j = t; j < L; j += blockDim.x) {
        float e = __expf((srow[j] - smax) / T);
        srow[j] = e;
        lsum += e;
    }
    red[t] = lsum;
    __syncthreads();
    for (int off = 128; off > 0; off >>= 1) {
        if (t < off) red[t] += red[t + off];
        __syncthreads();
    }
    float inv = 1.f / red[0];
    __syncthreads();
    for (int j = t; j < L; j += blockDim.x) {
        A_out[(size_t)i * L + j] = srow[j] * inv;
    }
}

// ---------------------------------------------------------------------------
// Kernel 3: fp32 WMMA GEMM with TDM-staged, double-buffered LDS tiles.
//   C[M,N] = A[M,K] * B (+bias)
//   BT=true : B stored [N,K] row-major (weights; C = A*B^T)
//   BT=false: B stored [K,N] row-major (A@V)
// Block = 256 thr = 8 waves (2 M x 4 N), 128x128 C tile, K-slab = 32.
// TDM (wave 0) DMAs A-panel 128x32 and B-panel into LDS; waves ds_load
// fragments. A/W tiles padded to pitch 34 (TDM pad_interval=32DW, pad=2DW)
// for conflict-free b64 LDS reads. Per-wave tile: 64x32 (4x2 16x16 accs).
// ---------------------------------------------------------------------------
#define KS 32
#define PA 34  // padded LDS pitch for 32-wide K-slab rows

template <bool BT>
__global__ __launch_bounds__(256) void wmma_gemm_f32_kernel(
    const float* __restrict__ A, const float* __restrict__ B,
    const float* __restrict__ bias, float* __restrict__ C,
    int M, int N, int K) {
    __shared__ float lsA[2][128 * PA];                       // 128 x 32 (pitch 34)
    __shared__ float lsB[2][BT ? (128 * PA) : (KS * 128)];   // 128x32 p34 | 32x128

    const int lane = threadIdx.x & 31;
    const int wave = threadIdx.x >> 5;
    const int wm = wave >> 2;   // 0..1
    const int wn = wave & 3;    // 0..3
    const int m0 = blockIdx.y * 128;
    const int n0 = blockIdx.x * 128;
    const int row = lane & 15;           // A: M row in tile; B: N col in tile
    const int kh = (lane >> 4) << 1;     // lanes 0-15 -> K+{0,1}; 16-31 -> K+{2,3}
    const int mbase = (lane >> 4) << 3;  // C rows 0..7 / 8..15

    const bool issuer = (wave == 0);
    const int nslab = K / KS;

    // TDM descriptors (uniform). group1 is slab-invariant; group0 moves.
    const unsigned long long gA0 = (unsigned long long)(uintptr_t)A +
                                   ((unsigned long long)m0 * K) * 4ull;
    unsigned long long gB0;
    u32x8 g1A = tdm_g1(K, 128, KS, 128, (unsigned long long)K, 1, 4, 1);
    u32x8 g1B;
    if (BT) {
        gB0 = (unsigned long long)(uintptr_t)B + ((unsigned long long)n0 * K) * 4ull;
        g1B = tdm_g1(K, 128, KS, 128, (unsigned long long)K, 1, 4, 1);
    } else {
        gB0 = (unsigned long long)(uintptr_t)B + (unsigned long long)n0 * 4ull;
        g1B = tdm_g1(N, KS, 128, KS, (unsigned long long)N, 0, 0, 0);
    }
    const unsigned ldsA0 = (unsigned)(uintptr_t)&lsA[0][0];
    const unsigned ldsA1 = (unsigned)(uintptr_t)&lsA[1][0];
    const unsigned ldsB0 = (unsigned)(uintptr_t)&lsB[0][0];
    const unsigned ldsB1 = (unsigned)(uintptr_t)&lsB[1][0];

    // Prologue: DMA slab 0 into buffer 0.
    if (issuer) {
        tdm_load_2d(tdm_g0(ldsA0, gA0), g1A);
        tdm_load_2d(tdm_g0(ldsB0, gB0), g1B);
        tdm_wait();
    }
    __syncthreads();

    v8f acc[4][2] = {};

    for (int s = 0; s < nslab; ++s) {
        const int cur = s & 1;
        // Kick off DMA for next slab into the other buffer.
        if (s + 1 < nslab && issuer) {
            unsigned long long offA = gA0 + (unsigned long long)(s + 1) * KS * 4ull;
            unsigned long long offB;
            if (BT)
                offB = gB0 + (unsigned long long)(s + 1) * KS * 4ull;
            else
                offB = gB0 + (unsigned long long)(s + 1) * KS * (unsigned long long)N * 4ull;
            tdm_load_2d(tdm_g0(cur ? ldsA0 : ldsA1, offA), g1A);
            tdm_load_2d(tdm_g0(cur ? ldsB0 : ldsB1, offB), g1B);
        }

        // Compute current slab from LDS.
        const float* As = &lsA[cur][0];
        const float* Bs = &lsB[cur][0];
        const int ar = wm * 64 + row;  // A row base for this wave
        const int nc = wn * 32 + row;  // B col base for this wave
#pragma unroll
        for (int kk = 0; kk < KS; kk += 4) {
            const int kkh = kk + kh;
            v2f a0, a1, a2, a3, b0, b1;
            a0.x = As[(ar + 0) * PA + kkh];  a0.y = As[(ar + 0) * PA + kkh + 1];
            a1.x = As[(ar + 16) * PA + kkh]; a1.y = As[(ar + 16) * PA + kkh + 1];
            a2.x = As[(ar + 32) * PA + kkh]; a2.y = As[(ar + 32) * PA + kkh + 1];
            a3.x = As[(ar + 48) * PA + kkh]; a3.y = As[(ar + 48) * PA + kkh + 1];
            if (BT) {
                b0.x = Bs[(nc + 0) * PA + kkh];  b0.y = Bs[(nc + 0) * PA + kkh + 1];
                b1.x = Bs[(nc + 16) * PA + kkh]; b1.y = Bs[(nc + 16) * PA + kkh + 1];
            } else {
                b0.x = Bs[kkh * 128 + nc + 0];        b0.y = Bs[(kkh + 1) * 128 + nc + 0];
                b1.x = Bs[kkh * 128 + nc + 16];       b1.y = Bs[(kkh + 1) * 128 + nc + 16];
            }
            acc[0][0] = __builtin_amdgcn_wmma_f32_16x16x4_f32(false, a0, false, b0, (short)0, acc[0][0], false, false);
            acc[0][1] = __builtin_amdgcn_wmma_f32_16x16x4_f32(false, a0, false, b1, (short)0, acc[0][1], false, false);
            acc[1][0] = __builtin_amdgcn_wmma_f32_16x16x4_f32(false, a1, false, b0, (short)0, acc[1][0], false, false);
            acc[1][1] = __builtin_amdgcn_wmma_f32_16x16x4_f32(false, a1, false, b1, (short)0, acc[1][1], false, false);
            acc[2][0] = __builtin_amdgcn_wmma_f32_16x16x4_f32(false, a2, false, b0, (short)0, acc[2][0], false, false);
            acc[2][1] = __builtin_amdgcn_wmma_f32_16x16x4_f32(false, a2, false, b1, (short)0, acc[2][1], false, false);
            acc[3][0] = __builtin_amdgcn_wmma_f32_16x16x4_f32(false, a3, false, b0, (short)0, acc[3][0], false, false);
            acc[3][1] = __builtin_amdgcn_wmma_f32_16x16x4_f32(false, a3, false, b1, (short)0, acc[3][1], false, false);
        }

        // Ensure next slab's DMA landed, then everyone flips buffers.
        if (issuer) tdm_wait();
        __syncthreads();
    }

    // Write back C (+bias).
#pragma unroll
    for (int nt = 0; nt < 2; nt++) {
        int n = n0 + wn * 32 + nt * 16 + row;
        float bv = bias ? bias[n] : 0.f;
#pragma unroll
        for (int mt = 0; mt < 4; mt++) {
            float* pC = C + (size_t)(m0 + wm * 64 + mt * 16 + mbase) * N + n;
#pragma unroll
            for (int r = 0; r < 8; r++) {
                pC[0] = acc[mt][nt][r] + bv;
                pC += N;
            }
        }
    }
}

// ---------------------------------------------------------------------------
// Launch
// ---------------------------------------------------------------------------
extern "C" void kernel_launch(void* const* d_in, const int* in_sizes, int n_in,
                              void* d_out, int out_size, void* d_ws, size_t ws_size,
                              hipStream_t stream) {
    const float* X      = (const float*)d_in[0];   // token_vecs (L,d)
    const int*   pos_i  = (const int*)d_in[1];     // pos_idx (L)
    const int*   depth  = (const int*)d_in[2];     // depth (L)
    const int*   slot   = (const int*)d_in[3];     // slot_idx (L)
    const float* tpos   = (const float*)d_in[4];   // token_pos (L)
    const float* conf   = (const float*)d_in[5];   // confidences (L)
    const float* res    = (const float*)d_in[6];   // resonances (L)
    const float* Gmic   = (const float*)d_in[7];   // G_micro (8,8)
    const float* Gmac   = (const float*)d_in[8];   // G_macro (1)
    const float* kspr   = (const float*)d_in[9];   // k_spring (8)
    const float* temp   = (const float*)d_in[10];  // temperature (1)
    const float* Wv_w   = (const float*)d_in[11];  // (d,d)
    const float* Wv_b   = (const float*)d_in[12];  // (d)
    const float* Wout_w = (const float*)d_in[13];  // (d,d)
    const float* Wout_b = (const float*)d_in[14];  // (d)

    const int L = in_sizes[1];
    const int d = in_sizes[0] / L;

    float* out    = (float*)d_out;              // (L,d)
    float* A      = out + (size_t)L * d;        // (L,L)
    float* scores = A + (size_t)L * L;          // (L,L)

    float* ws     = (float*)d_ws;
    float* masses = ws;
    float* bh     = ws + (size_t)L;
    float* kpi    = ws + 2 * (size_t)L;
    float* coll   = ws + 3 * (size_t)L;
    float* V      = ws + 4 * (size_t)L;         // (L,d)
    float* AV     = V + (size_t)L * d;          // (L,d)

    // 1. per-token stats
    token_stats_kernel<<<L, 256, 0, stream>>>(X, pos_i, conf, res, kspr,
                                              masses, bh, kpi, coll, d);

    // 2. scores + softmax (row-resident in LDS)
    scores_softmax_kernel<<<L, 256, (size_t)L * sizeof(float), stream>>>(
        masses, bh, kpi, coll, pos_i, depth, slot, tpos, Gmic, Gmac, temp,
        scores, A, L);

    // 3. V = X @ Wv^T + Wv_b
    dim3 grid_g(d / 128, L / 128);
    wmma_gemm_f32_kernel<true><<<grid_g, 256, 0, stream>>>(X, Wv_w, Wv_b, V, L, d, d);

    // 4. AV = A @ V
    wmma_gemm_f32_kernel<false><<<grid_g, 256, 0, stream>>>(A, V, nullptr, AV, L, d, L);

    // 5. out = AV @ Wout^T + Wout_b
    wmma_gemm_f32_kernel<true><<<grid_g, 256, 0, stream>>>(AV, Wout_w, Wout_b, out, L, d, d);
}